// ELL_7206955122817
// MI455X (gfx1250) — compile-verified
//
#include <hip/hip_runtime.h>
#include <math.h>

// MI455X (gfx1250) — wave32. One workgroup (256 thr = 8 waves) per row.
// Bandwidth-bound: 128 MiB of samples @ 23.3 TB/s ≈ 5.5 us floor; per-element
// math reduced to the hardware minimum of 4 TRANS ops (exp2/log2 folding).

typedef __attribute__((ext_vector_type(2))) float v2f;
typedef __attribute__((ext_vector_type(8))) float v8f;

#define LOG2E_F   1.4426950408889634f
#define TWO_PI_F  6.2831853071795865f

__global__ __launch_bounds__(256) void ell_row_kernel(
    const float* __restrict__ samples,
    const int*   __restrict__ y_true,
    const float* __restrict__ machine_mu,
    const float* __restrict__ machine_var,
    const float* __restrict__ user_theta,
    const float* __restrict__ encoder_mu,
    float* __restrict__ g_out,   // [B] gaussian_loss per row
    float* __restrict__ b_out,   // [B] beta_loss per row
    float* __restrict__ p_out,   // [B] precision_loss per row
    int S)
{
    const int row  = blockIdx.x;
    const int tid  = threadIdx.x;
    const unsigned lane = tid & 31u;
    const int wave = tid >> 5;

    const float mu  = machine_mu[row];
    const float var = machine_var[row];
    const float a   = user_theta[row * 2 + 1];   // theta[:,1]
    const float b   = user_theta[row * 2 + 0];   // theta[:,0]

    const float cg  = -0.5f * LOG2E_F / var;     // exp(-.5 t^2/var) = exp2(cg*t*t)
    const float am1 = a - 1.0f;
    const float bm1 = b - 1.0f;

    // --- Streaming phase: coalesced float4 loads, per-lane partial sums ---
    const float4* __restrict__ xrow = (const float4*)(samples + (size_t)row * (size_t)S);
    const int n4 = S >> 2;

    float gsum = 0.0f;
    float bsum = 0.0f;
    for (int j = tid; j < n4; j += 256) {
        float4 x4 = xrow[j];
        float xs[4] = {x4.x, x4.y, x4.z, x4.w};
#pragma unroll
        for (int k = 0; k < 4; ++k) {
            float xv = xs[k];
            float t  = xv - mu;
            gsum += __builtin_amdgcn_exp2f(cg * t * t);                 // v_exp_f32
            float lx = __builtin_amdgcn_logf(xv);                       // v_log_f32 (log2)
            float l1 = __builtin_amdgcn_logf(1.0f - xv);                // v_log_f32
            bsum += __builtin_amdgcn_exp2f(fmaf(am1, lx, bm1 * l1));    // v_exp_f32
        }
    }

    // --- In-wave reduction of (gsum, bsum) via V_WMMA_F32_16X16X4_F32 ---
    // A (16x4 f32): vgpr0 = K0 (lanes 0-15) / K2 (lanes 16-31); vgpr1 = K1/K3.
    //   A[m][0]=gsum(lane m)  A[m][1]=bsum(lane m)
    //   A[m][2]=gsum(lane m+16) A[m][3]=bsum(lane m+16)
    // B (4x16 f32): B[0][0]=B[2][0]=1 and B[1][1]=B[3][1]=1, else 0
    //   => D[m][0] = gsum_m + gsum_{m+16},  D[m][1] = bsum_m + bsum_{m+16}
    v2f amat;
    amat[0] = gsum;
    amat[1] = bsum;
    v2f bmat;
    bmat[0] = (lane == 0u || lane == 16u) ? 1.0f : 0.0f;  // rows K0,K2 -> col N=0
    bmat[1] = (lane == 1u || lane == 17u) ? 1.0f : 0.0f;  // rows K1,K3 -> col N=1
    v8f cmat = {};
    cmat = __builtin_amdgcn_wmma_f32_16x16x4_f32(
        /*neg_a=*/false, amat, /*neg_b=*/false, bmat,
        /*c_mod=*/(short)0, cmat, /*reuse_a=*/false, /*reuse_b=*/false);

    // Sum the 8 D VGPRs: lane0 = sum_{m=0..7} D[m][0], lane16 = sum_{m=8..15} D[m][0]
    //                    lane1 / lane17 likewise for column 1 (bsum).
    float s = cmat[0] + cmat[1] + cmat[2] + cmat[3] +
              cmat[4] + cmat[5] + cmat[6] + cmat[7];
    float tot = s + __shfl_xor(s, 16, 32);   // lane0: wave gsum, lane1: wave bsum
    float wave_g = __shfl(tot, 0, 32);
    float wave_b = __shfl(tot, 1, 32);

    // --- Cross-wave reduction (8 waves) + per-row epilogue ---
    __shared__ float sg[8];
    __shared__ float sb[8];
    if (lane == 0u) { sg[wave] = wave_g; sb[wave] = wave_b; }
    __syncthreads();

    if (tid == 0) {
        float G = 0.0f, Bt = 0.0f;
#pragma unroll
        for (int w = 0; w < 8; ++w) { G += sg[w]; Bt += sb[w]; }
        const float invS = 1.0f / (float)S;

        // gaussian_loss = exp(-mean(exp(g_logp))) ; C_row = 1/sqrt(2*pi*var)
        float c_row = rsqrtf(TWO_PI_F * var);
        float gl = expf(-(G * invS) * c_row);

        // beta_loss: factor exp(-lbeta) out of the sum
        float lbeta = lgammaf(a) + lgammaf(b) - lgammaf(a + b);
        float bl = expf(-(Bt * invS) * expf(-lbeta));

        // precision_loss = logaddexp(0,z) - z*y
        float z = encoder_mu[row];
        float y = (float)y_true[row];
        float pl = fmaxf(z, 0.0f) + log1pf(expf(-fabsf(z))) - z * y;

        g_out[row] = gl;
        b_out[row] = bl;
        p_out[row] = pl;
    }
}

// Deterministic final means over B rows (single block, fixed tree order).
__global__ __launch_bounds__(1024) void ell_reduce_kernel(
    const float* __restrict__ g_in,
    const float* __restrict__ b_in,
    const float* __restrict__ p_in,
    float* __restrict__ out, int B)
{
    __shared__ float s0[1024];
    __shared__ float s1[1024];
    __shared__ float s2[1024];
    const int tid = threadIdx.x;
    float ag = 0.0f, ab = 0.0f, ap = 0.0f;
    for (int i = tid; i < B; i += 1024) {
        ag += g_in[i]; ab += b_in[i]; ap += p_in[i];
    }
    s0[tid] = ag; s1[tid] = ab; s2[tid] = ap;
    __syncthreads();
    for (int off = 512; off > 0; off >>= 1) {
        if (tid < off) {
            s0[tid] += s0[tid + off];
            s1[tid] += s1[tid + off];
            s2[tid] += s2[tid + off];
        }
        __syncthreads();
    }
    if (tid == 0) {
        float invB = 1.0f / (float)B;
        float mg = s0[0] * invB, mb = s1[0] * invB, mp = s2[0] * invB;
        out[0] = (s0[0] + s1[0] + s2[0]) * invB;  // mean(loss)
        out[1] = mg;
        out[2] = mb;
        out[3] = mp;
    }
}

extern "C" void kernel_launch(void* const* d_in, const int* in_sizes, int n_in,
                              void* d_out, int out_size, void* d_ws, size_t ws_size,
                              hipStream_t stream) {
    const float* samples     = (const float*)d_in[0];
    const int*   y_true      = (const int*)  d_in[1];
    const float* machine_mu  = (const float*)d_in[2];
    const float* machine_var = (const float*)d_in[3];
    const float* user_theta  = (const float*)d_in[4];
    const float* encoder_mu  = (const float*)d_in[5];

    const int B = in_sizes[1];              // y_true is [B]
    const int S = in_sizes[0] / B;          // samples is [B,S]

    float* gw = (float*)d_ws;               // [B]
    float* bw = gw + B;                     // [B]
    float* pw = gw + 2 * B;                 // [B]

    ell_row_kernel<<<B, 256, 0, stream>>>(samples, y_true, machine_mu, machine_var,
                                          user_theta, encoder_mu, gw, bw, pw, S);
    ell_reduce_kernel<<<1, 1024, 0, stream>>>(gw, bw, pw, (float*)d_out, B);
}